// StyleTextToSpeech2HarmonicNoiseSourceFilter_14396730377034
// MI455X (gfx1250) — compile-verified
//
#include <hip/hip_runtime.h>
#include <math.h>

// ---- problem constants (from reference) ------------------------------------
#define NSF_B     8
#define NSF_T     614400
#define NSF_C     9          // H + 1 harmonics
#define NSF_US    300        // upsample scale
#define NSF_DLEN  2048       // T / US
#define NSF_AMP   0.1f
#define NSF_NSTD  0.003f

typedef __attribute__((ext_vector_type(2))) float v2f;
typedef __attribute__((ext_vector_type(8))) float v8f;

// ---------------------------------------------------------------------------
// Kernel 1: build the per-(b,c) phase table (cumsum of downsampled rad), in
// "turns" (multiply by 2*pi at use).  The /300 downsample with
// align_corners=False samples exactly t=300i+149 and t=300i+150 with w=0.5,
// so rand_ini (added at t=0 only) never reaches the output and is skipped.
// One block per (b,c) pair: 72 blocks x 256 threads, 8 elements/thread,
// LDS Hillis-Steele scan for the block prefix.
// ---------------------------------------------------------------------------
__global__ __launch_bounds__(256)
void nsf_phase_kernel(const float* __restrict__ f0, float* __restrict__ phase) {
    const int pair = blockIdx.x;            // 0..71
    const int b = pair / NSF_C;
    const int c = pair % NSF_C;
    const float hc = (float)(c + 1);
    const float inv_sr = 1.0f / 24000.0f;
    const int tid = threadIdx.x;

    const float* f0b = f0 + (size_t)b * NSF_T;

    float r[8];
    float run = 0.0f;
#pragma unroll
    for (int k = 0; k < 8; ++k) {
        const int e  = tid * 8 + k;          // downsampled index, 0..2047
        const int ta = e * NSF_US + 149;     // lo sample;  ta+1 = hi sample
        float xa = f0b[ta]     * hc * inv_sr;
        float xb = f0b[ta + 1] * hc * inv_sr;
        xa -= floorf(xa);                    // rad % 1.0
        xb -= floorf(xb);
        run += 0.5f * (xa + xb);             // lerp with w = 0.5
        r[k] = run;                          // local inclusive running sum
    }

    __shared__ float sdata[256];
    const float total = run;
    sdata[tid] = total;
    __syncthreads();
#pragma unroll
    for (int off = 1; off < 256; off <<= 1) {
        const float v = (tid >= off) ? sdata[tid - off] : 0.0f;
        __syncthreads();
        sdata[tid] += v;
        __syncthreads();
    }
    const float excl = sdata[tid] - total;   // exclusive prefix of this thread

    float* pout = phase + (size_t)pair * NSF_DLEN + tid * 8;
#pragma unroll
    for (int k = 0; k < 8; ++k) pout[k] = excl + r[k];   // cumsum, in turns
}

// ---------------------------------------------------------------------------
// Kernel 2: streaming output kernel.  Each wave produces 16 outputs via three
// chained V_WMMA_F32_16X16X4_F32 ops (channels 0..11, 9 real + 3 zero pad).
//
//   A (16x4)  : weight chunk broadcast across all M rows.
//               x4-f32 operand layout: V0 = K=(half?2:0), V1 = K=(half?3:1).
//   B (4x16)  : per-timestep signal x[c][t], N = lane%16, same K-per-half map.
//   D (16x16) : every M-row identical (A rows identical), so acc[0] of every
//               lane is the dot product for timestep t_base + lane%16.
//
// Per-lane channel work (interp + sin + noise mix) splits across wave halves.
// ---------------------------------------------------------------------------
__global__ __launch_bounds__(256)
void nsf_main_kernel(const float* __restrict__ f0,
                     const float* __restrict__ noise,
                     const float* __restrict__ weight,
                     const float* __restrict__ bias,
                     const float* __restrict__ phase,
                     float* __restrict__ out) {
    const int tid  = threadIdx.x;
    const int wave = tid >> 5;
    const int lane = tid & 31;
    const int half = lane >> 4;     // which K-pair this lane supplies
    const int tt   = lane & 15;     // N column == timestep within the group

    const int blocksPerBatch = NSF_T / 128;              // 4800
    const int b    = blockIdx.x / blocksPerBatch;
    const int tblk = (blockIdx.x % blocksPerBatch) * 128;
    const int t    = tblk + wave * 16 + tt;

    const float f0v  = f0[(size_t)b * NSF_T + t];
    const float uv   = (f0v > 0.0f) ? 1.0f : 0.0f;
    const float samp = NSF_AMP * uv;                             // sine amp * uv
    const float namp = uv * NSF_NSTD + (1.0f - uv) * (NSF_AMP / 3.0f);

    // upsample coords: (t+0.5)/300 - 0.5, clipped to [0, 2047]
    float fc = ((float)t + 0.5f) * (1.0f / (float)NSF_US) - 0.5f;
    fc = fminf(fmaxf(fc, 0.0f), (float)(NSF_DLEN - 1));
    const int   lo = (int)floorf(fc);
    const int   hi = min(lo + 1, NSF_DLEN - 1);
    const float w  = fc - (float)lo;

    const float* ph = phase + (size_t)b * NSF_C * NSF_DLEN;
    const float* nz = noise + ((size_t)b * NSF_T + t) * NSF_C;
    const float TWO_PI_US = 6.2831853071795864769f * (float)NSF_US;

    v8f acc = {};
#pragma unroll
    for (int k = 0; k < 3; ++k) {
        v2f a, bb;
#pragma unroll
        for (int j = 0; j < 2; ++j) {
            const int c = 4 * k + 2 * half + j;     // channel fed by this reg
            float aval = 0.0f, xval = 0.0f;
            if (c < NSF_C) {
                aval = weight[c];
                const float* pc = ph + c * NSF_DLEN;
                const float p = pc[lo];
                const float q = pc[hi];
                const float ang = (p + w * (q - p)) * TWO_PI_US; // phase [rad]
                xval = __sinf(ang) * samp + namp * nz[c];
            }
            a[j]  = aval;
            bb[j] = xval;
        }
        // D = A x B + C   (chained accumulation over K-chunks)
        acc = __builtin_amdgcn_wmma_f32_16x16x4_f32(
            /*neg_a=*/false, a, /*neg_b=*/false, bb,
            /*c_mod=*/(short)0, acc, /*reuse_a=*/false, /*reuse_b=*/false);
    }

    const float y = tanhf(acc[0] + bias[0]);
    if (half == 0) {                      // 16 unique results per wave
        out[(size_t)b * NSF_T + t] = y;
    }
}

// ---------------------------------------------------------------------------
extern "C" void kernel_launch(void* const* d_in, const int* in_sizes, int n_in,
                              void* d_out, int out_size, void* d_ws, size_t ws_size,
                              hipStream_t stream) {
    (void)in_sizes; (void)n_in; (void)out_size; (void)ws_size;

    const float* f0     = (const float*)d_in[0];   // [B, T, 1]
    // d_in[1] = rand_ini: provably unused (t=0 never sampled by downsample)
    const float* noise  = (const float*)d_in[2];   // [B, T, C]
    const float* weight = (const float*)d_in[3];   // [1, C]
    const float* bias   = (const float*)d_in[4];   // [1]
    float* out   = (float*)d_out;                  // [B, T, 1]
    float* phase = (float*)d_ws;                   // [B*C, 2048] scratch

    nsf_phase_kernel<<<NSF_B * NSF_C, 256, 0, stream>>>(f0, phase);

    const int nblocks = (NSF_B * NSF_T) / 128;     // 38400 blocks, 16 out/wave
    nsf_main_kernel<<<nblocks, 256, 0, stream>>>(f0, noise, weight, bias,
                                                 phase, out);
}